// MultiModelARIX_16432544875253
// MI455X (gfx1250) — compile-verified
//
#include <hip/hip_runtime.h>

// Problem constants (match reference)
#define BATCH 4096
#define SEQ   2048
#define ARP   8
#define EXO   8
#define USTR  (SEQ + EXO)      // 2056 floats per batch row of u
#define OSTR  (SEQ + ARP + 1)  // 2057 floats per batch row of out
#define KTAPS 256              // truncated impulse-response length (|h| < fp32 eps beyond)
#define EROWS (SEQ + 8)        // live Efull rows: 8 virtual-init rows + SEQ exo rows
#define EROWS_PAD (EROWS + 8)  // + zeroed pad rows so band-edge loads are unconditional
#define HOFF  32               // h lives at hpad[HOFF .. HOFF+KTAPS)
#define HPAD_LEN 512           // zeroed; covers [HOFF-18, HOFF+KTAPS+18) with margin

typedef __attribute__((ext_vector_type(2))) float v2f;
typedef __attribute__((ext_vector_type(8))) float v8f;

// ---------------------------------------------------------------------------
// 1) Impulse response of the difference process z: h[n] = sum_d W[8-d] h[n-d],
//    h[0]=1, written into a zero-padded buffer (pad makes GEMM loads branch-free
//    and makes the recurrence itself read zeros for n-d < 0).
// ---------------------------------------------------------------------------
__global__ void k_impulse(const float* __restrict__ W, float* __restrict__ hpad) {
  if (threadIdx.x == 0 && blockIdx.x == 0) {
    for (int i = 0; i < HPAD_LEN; ++i) hpad[i] = 0.0f;
    hpad[HOFF] = 1.0f;
    for (int n = 1; n < KTAPS; ++n) {
      float s = 0.0f;
#pragma unroll
      for (int d = 1; d <= 8; ++d) s += W[8 - d] * hpad[HOFF + n - d];
      hpad[HOFF + n] = s;
    }
  }
}

// ---------------------------------------------------------------------------
// 2) Fold initial-window diffs into 8 virtual inputs eps (unit-lower-triangular
//    solve against h). Also zeroes the Efull pad rows (workspace is poisoned).
// ---------------------------------------------------------------------------
__global__ void k_eps(const float* __restrict__ y, const float* __restrict__ hpad,
                      float* __restrict__ Efull) {
  const int b = blockIdx.x * blockDim.x + threadIdx.x;
  if (b >= BATCH) return;
  float hv[8];
#pragma unroll
  for (int i = 0; i < 8; ++i) hv[i] = hpad[HOFF + i];
  float e[8];
#pragma unroll
  for (int i = 0; i < 8; ++i) {
    float s = y[b * 9 + i + 1] - y[b * 9 + i];  // z_{i+1}
#pragma unroll
    for (int p = 0; p < 8; ++p) {
      if (p < i) s -= hv[i - p] * e[p];
    }
    e[i] = s;
    Efull[i * BATCH + b] = s;
  }
#pragma unroll
  for (int i = 0; i < 8; ++i)
    Efull[(size_t)(EROWS + i) * BATCH + b] = 0.0f;  // zero pad rows
}

// ---------------------------------------------------------------------------
// 3) Exogenous drive e[m,b] = sum_j W[8+j] * u[b, 1+m+j].  Layout [m][b]
//    (b fastest) so GEMM B-operand loads are coalesced. u stays L2-hot.
// ---------------------------------------------------------------------------
__global__ void k_exo(const float* __restrict__ u, const float* __restrict__ W,
                      float* __restrict__ Efull) {
  const int b = blockIdx.x * blockDim.x + threadIdx.x;  // gridDim.x = BATCH/256
  const int m = blockIdx.y;                             // 0..SEQ-1
  float wv[8];
#pragma unroll
  for (int j = 0; j < 8; ++j) wv[j] = W[8 + j];
  const float* up = u + (size_t)b * USTR + 1 + m;
  float s = 0.0f;
#pragma unroll
  for (int j = 0; j < 8; ++j) s += wv[j] * up[j];
  Efull[(size_t)(m + 8) * BATCH + b] = s;
}

// ---------------------------------------------------------------------------
// 4) Banded Toeplitz GEMM on WMMA:  Z[n,b] = sum_r hpad[HOFF + (n+7) - r] * Efull[r,b]
//    Each wave: one 16(n) x 64(b) tile = four independent f32 accumulators
//    sharing the A fragment (one b64 load of h per K-step). Branch-free band
//    edges via zero padding; B loads are coalesced row segments of Efull.
// ---------------------------------------------------------------------------
__global__ void k_zconv(const float* __restrict__ hpad, const float* __restrict__ Efull,
                        float* __restrict__ Z) {
  const int lane = threadIdx.x & 31;
  const int wv   = threadIdx.x >> 5;          // wave 0..3 in block
  const int nbase = blockIdx.x * 16 + 1;      // smallest n of this tile
  const int b0 = blockIdx.y * 256 + wv * 64;  // batch tile base (64 cols/wave)
  const int col  = lane & 15;                 // M for A, N for B/D
  const int half = lane >> 4;                 // lane half selects K pair / D rows

  v8f acc0 = {0.0f, 0.0f, 0.0f, 0.0f, 0.0f, 0.0f, 0.0f, 0.0f};
  v8f acc1 = {0.0f, 0.0f, 0.0f, 0.0f, 0.0f, 0.0f, 0.0f, 0.0f};
  v8f acc2 = {0.0f, 0.0f, 0.0f, 0.0f, 0.0f, 0.0f, 0.0f, 0.0f};
  v8f acc3 = {0.0f, 0.0f, 0.0f, 0.0f, 0.0f, 0.0f, 0.0f, 0.0f};

  int r0 = nbase + 8 - KTAPS;                 // first r with h-index < KTAPS
  if (r0 < 0) r0 = 0;
  r0 &= ~3;
  const int rmax = nbase + 22;                // last r with h-index >= 0 in tile

  // A element (M=col, K=ka): hpad[HOFF + (nbase+col) + 7 - ka], ka = r + 2*half
  const float* hbase = hpad + (HOFF + nbase + col + 7 - 2 * half - r0);
  // B element (K=ka, N): Efull[(r + 2*half) * BATCH + b0 + col + 16*frag]
  const float* ebase = Efull + ((size_t)(r0 + 2 * half) * BATCH + b0 + col);

  for (int r = r0; r <= rmax; r += 4) {
    v2f a;
    a.x = hbase[0];
    a.y = hbase[-1];
    v2f f0, f1, f2, f3;
    f0.x = ebase[0];          f0.y = ebase[BATCH];
    f1.x = ebase[16];         f1.y = ebase[BATCH + 16];
    f2.x = ebase[32];         f2.y = ebase[BATCH + 32];
    f3.x = ebase[48];         f3.y = ebase[BATCH + 48];
    acc0 = __builtin_amdgcn_wmma_f32_16x16x4_f32(false, a, false, f0, (short)0,
                                                 acc0, false, false);
    acc1 = __builtin_amdgcn_wmma_f32_16x16x4_f32(false, a, false, f1, (short)0,
                                                 acc1, false, false);
    acc2 = __builtin_amdgcn_wmma_f32_16x16x4_f32(false, a, false, f2, (short)0,
                                                 acc2, false, false);
    acc3 = __builtin_amdgcn_wmma_f32_16x16x4_f32(false, a, false, f3, (short)0,
                                                 acc3, false, false);
    hbase -= 4;
    ebase += 4 * BATCH;
  }

#pragma unroll
  for (int v = 0; v < 8; ++v) {
    const int row = v + 8 * half;             // D layout: VGPR v -> M = v + 8*half
    float* zp = Z + ((size_t)(nbase - 1 + row) * BATCH + b0 + col);
    zp[0]  = acc0[v];
    zp[16] = acc1[v];
    zp[32] = acc2[v];
    zp[48] = acc3[v];
  }
}

// ---------------------------------------------------------------------------
// 5) Integrate: x_{8+n} = x_8 + cumsum(z). Coalesced across 4096 lanes.
// ---------------------------------------------------------------------------
__global__ void k_cumsum(const float* __restrict__ y, const float* __restrict__ Z,
                         float* __restrict__ out) {
  const int b = blockIdx.x * blockDim.x + threadIdx.x;
  if (b >= BATCH) return;
  float* ob = out + (size_t)b * OSTR;
#pragma unroll
  for (int k = 0; k < 9; ++k) ob[k] = y[b * 9 + k];
  float x = y[b * 9 + 8];
  for (int t = 0; t < SEQ; ++t) {
    x += Z[(size_t)t * BATCH + b];
    ob[9 + t] = x;
  }
}

// ---------------------------------------------------------------------------
extern "C" void kernel_launch(void* const* d_in, const int* in_sizes, int n_in,
                              void* d_out, int out_size, void* d_ws, size_t ws_size,
                              hipStream_t stream) {
  (void)in_sizes; (void)n_in; (void)out_size; (void)ws_size;
  const float* y = (const float*)d_in[0];  // [B,1,9]
  const float* u = (const float*)d_in[1];  // [B,1,2056]
  const float* W = (const float*)d_in[2];  // [1,16]
  float* out = (float*)d_out;              // [B,1,2057]

  char* ws = (char*)d_ws;
  float* hpad  = (float*)ws;                                           // HPAD_LEN floats
  float* Efull = (float*)(ws + HPAD_LEN * 4);                          // EROWS_PAD x BATCH
  float* Z     = (float*)(ws + HPAD_LEN * 4 +
                          (size_t)EROWS_PAD * BATCH * 4);              // SEQ x BATCH

  k_impulse<<<1, 32, 0, stream>>>(W, hpad);
  k_eps   <<<BATCH / 256, 256, 0, stream>>>(y, hpad, Efull);
  k_exo   <<<dim3(BATCH / 256, SEQ), 256, 0, stream>>>(u, W, Efull);
  k_zconv <<<dim3(SEQ / 16, BATCH / 256), 128, 0, stream>>>(hpad, Efull, Z);
  k_cumsum<<<BATCH / 256, 256, 0, stream>>>(y, Z, out);
}